// PAAPostProcessor_40965398069545
// MI455X (gfx1250) — compile-verified
//
#include <hip/hip_runtime.h>
#include <stdint.h>

// ---------------- problem constants ----------------
#define N_IMG   8
#define C_CLS   80
#define H_DIM   200
#define W_DIM   304
#define HW      (H_DIM * W_DIM)          /* 60800 */
#define K_TOP   10000
#define BINS    8192
#define SORT_N  16384                    /* next pow2 >= K_TOP */
#define TILE_HWA 1024
#define TILES   ((HW + TILE_HWA - 1) / TILE_HWA)   /* 60 */

#define THRESH_LOGIT (-2.94443897916644046f)  /* log(0.05/0.95): sigmoid(x)>0.05 */
#define BBOX_CLIP     (4.13516655674235627f)  /* log(1000/16) */
#define IMG_W_M1      2431.0f
#define IMG_H_M1      1599.0f

__device__ __forceinline__ float sigf(float x) {
    return 1.0f / (1.0f + __expf(-x));
}

__device__ __forceinline__ int score_bin(float s) {
    int b = (int)(s * (float)BINS);
    return b < 0 ? 0 : (b > BINS - 1 ? BINS - 1 : b);
}

// Low 32 bits of the LDS-aperture flat address == wave-relative LDS byte offset
// (ISA 10.2).
__device__ __forceinline__ unsigned lds_byte_off(const void* p) {
    return (unsigned)(uintptr_t)p;
}

// Issue one per-lane 16B async DMA from global address saddr plus voff to LDS byte offset ldsoff
__device__ __forceinline__ void async_ld_b128(unsigned ldsoff, unsigned voff,
                                              const void* saddr) {
    unsigned long long gb = (unsigned long long)(uintptr_t)saddr;
    asm volatile("global_load_async_to_lds_b128 %0, %1, %2"
                 :: "v"(ldsoff), "v"(voff), "s"(gb) : "memory");
}

// ---------------- kernel 0: zero histograms + counters ----------------
__global__ void k_zero(unsigned* __restrict__ p, int n) {
    int i = blockIdx.x * blockDim.x + threadIdx.x;
    if (i < n) p[i] = 0u;
}

// ---------------- kernel 1: fused sigmoid score + per-image histogram ------
// grid = (TILES, N_IMG), block = 256. Each thread owns 4 anchor positions.
// Class logits are streamed through a double-buffered LDS staging pipeline
// using CDNA5 async loads (ASYNCcnt), overlapping DMA of tile c+1 with the
// scoring of tile c. IoU sigmoids are computed once and reused for all 80
// classes from registers.
__global__ __launch_bounds__(256) void k_hist(
    const float* __restrict__ cls, const float* __restrict__ iou,
    unsigned* __restrict__ hist)
{
    __shared__ unsigned lhist[BINS];
    __shared__ float stage[2][TILE_HWA];

    const int tid  = threadIdx.x;
    const int tile = blockIdx.x;
    const int n    = blockIdx.y;

    for (int b = tid; b < BINS; b += 256) lhist[b] = 0u;
    __syncthreads();

    const int hwa0   = tile * TILE_HWA + tid * 4;
    const int hwa_cl = hwa0 > HW - 4 ? HW - 4 : hwa0;   // clamp OOB tail tile

    float4 iv = *(const float4*)(iou + (size_t)n * HW + hwa_cl);
    const float si[4] = {sigf(iv.x), sigf(iv.y), sigf(iv.z), sigf(iv.w)};

    const unsigned lds0 = lds_byte_off(&stage[0][tid * 4]);
    const unsigned lds1 = lds_byte_off(&stage[1][tid * 4]);
    const unsigned voff = (unsigned)(hwa_cl * 4);        // per-lane byte offset

    const float* base_n = cls + (size_t)n * C_CLS * HW;

    async_ld_b128(lds0, voff, base_n);                   // prologue: tile c=0

    #pragma unroll 2
    for (int c = 0; c < C_CLS; ++c) {
        if (c + 1 < C_CLS) {
            asm volatile("s_wait_dscnt 0x0" ::: "memory");   // prior LDS reads drained
            async_ld_b128(((c + 1) & 1) ? lds1 : lds0, voff,
                          base_n + (size_t)(c + 1) * HW);
            asm volatile("s_wait_asynccnt 0x1" ::: "memory"); // tile c landed
        } else {
            asm volatile("s_wait_asynccnt 0x0" ::: "memory");
        }

        float4 cv = *(const float4*)&stage[c & 1][tid * 4];
        const float x[4] = {cv.x, cv.y, cv.z, cv.w};
        #pragma unroll
        for (int j = 0; j < 4; ++j) {
            int hwa = hwa0 + j;
            if (hwa < HW && x[j] > THRESH_LOGIT) {
                float s = sqrtf(sigf(x[j]) * si[j]);
                atomicAdd(&lhist[score_bin(s)], 1u);
            }
        }
    }

    __syncthreads();
    unsigned* gh = hist + (size_t)n * BINS;
    for (int b = tid; b < BINS; b += 256) {
        unsigned v = lhist[b];
        if (v) atomicAdd(&gh[b], v);
    }
}

// ---------------- kernel 2: find K-th score threshold bin per image --------
__global__ void k_select(const unsigned* __restrict__ hist, int* __restrict__ meta) {
    const int n = blockIdx.x;
    if (threadIdx.x != 0) return;
    const unsigned* h = hist + (size_t)n * BINS;
    unsigned acc = 0, c_above = 0, remaining = 0, filled = 0;
    int B = -1;
    for (int b = BINS - 1; b >= 0; --b) {
        unsigned cb = h[b];
        if (acc + cb >= (unsigned)K_TOP) {
            B = b; c_above = acc; remaining = (unsigned)K_TOP - acc; break;
        }
        acc += cb;
    }
    filled = (B < 0) ? acc : (unsigned)K_TOP;   // B<0: fewer than K candidates
    int* m = meta + n * 4;
    m[0] = B; m[1] = (int)c_above; m[2] = (int)remaining; m[3] = (int)filled;
}

// ---------------- kernel 3: compaction pass (recompute scores) -------------
// Same tiled async-pipeline structure as k_hist; bin math is bit-identical so
// the per-bin counts match the histogram exactly. Winners (~K per image out of
// 4.86M) take slots via rare global atomics.
__global__ __launch_bounds__(256) void k_compact(
    const float* __restrict__ cls, const float* __restrict__ iou,
    const int* __restrict__ meta, unsigned* __restrict__ cnt,
    float* __restrict__ cscore, int* __restrict__ cidx)
{
    __shared__ float stage[2][TILE_HWA];

    const int tid  = threadIdx.x;
    const int tile = blockIdx.x;
    const int n    = blockIdx.y;

    const int* m = meta + n * 4;
    const int B = m[0];
    const unsigned c_above   = (unsigned)m[1];
    const unsigned remaining = (unsigned)m[2];

    const int hwa0   = tile * TILE_HWA + tid * 4;
    const int hwa_cl = hwa0 > HW - 4 ? HW - 4 : hwa0;

    float4 iv = *(const float4*)(iou + (size_t)n * HW + hwa_cl);
    const float si[4] = {sigf(iv.x), sigf(iv.y), sigf(iv.z), sigf(iv.w)};

    const unsigned lds0 = lds_byte_off(&stage[0][tid * 4]);
    const unsigned lds1 = lds_byte_off(&stage[1][tid * 4]);
    const unsigned voff = (unsigned)(hwa_cl * 4);

    const float* base_n = cls + (size_t)n * C_CLS * HW;

    async_ld_b128(lds0, voff, base_n);

    #pragma unroll 2
    for (int c = 0; c < C_CLS; ++c) {
        if (c + 1 < C_CLS) {
            asm volatile("s_wait_dscnt 0x0" ::: "memory");
            async_ld_b128(((c + 1) & 1) ? lds1 : lds0, voff,
                          base_n + (size_t)(c + 1) * HW);
            asm volatile("s_wait_asynccnt 0x1" ::: "memory");
        } else {
            asm volatile("s_wait_asynccnt 0x0" ::: "memory");
        }

        float4 cv = *(const float4*)&stage[c & 1][tid * 4];
        const float x[4] = {cv.x, cv.y, cv.z, cv.w};
        #pragma unroll
        for (int j = 0; j < 4; ++j) {
            int hwa = hwa0 + j;
            if (hwa < HW && x[j] > THRESH_LOGIT) {
                float s = sqrtf(sigf(x[j]) * si[j]);
                int b = score_bin(s);
                if (b > B) {
                    unsigned pos = atomicAdd(&cnt[n * 2 + 0], 1u); // pos < c_above
                    cscore[(size_t)n * K_TOP + pos] = s;
                    cidx  [(size_t)n * K_TOP + pos] = hwa * C_CLS + c;
                } else if (b == B) {
                    unsigned t = atomicAdd(&cnt[n * 2 + 1], 1u);
                    if (t < remaining) {
                        unsigned pos = c_above + t;
                        cscore[(size_t)n * K_TOP + pos] = s;
                        cidx  [(size_t)n * K_TOP + pos] = hwa * C_CLS + c;
                    }
                }
            }
        }
    }
}

// ---------------- kernel 4: LDS bitonic sort + decode + output -------------
// One 1024-thread block per image; 128 KB dynamic LDS (possible only with
// CDNA5's 320 KB-per-WGP LDS).
__global__ __launch_bounds__(1024) void k_sort_out(
    const float* __restrict__ reg, const float* __restrict__ anchors,
    const int* __restrict__ meta,
    const float* __restrict__ cscore, const int* __restrict__ cidx,
    float* __restrict__ out)
{
    extern __shared__ char smem[];
    float* key = (float*)smem;
    int*   val = (int*)(smem + (size_t)SORT_N * sizeof(float));

    const int n = blockIdx.x;
    const int tid = threadIdx.x;
    const int filled = meta[n * 4 + 3];

    for (int i = tid; i < SORT_N; i += 1024) {
        if (i < filled) {
            key[i] = cscore[(size_t)n * K_TOP + i];
            val[i] = cidx [(size_t)n * K_TOP + i];
        } else {
            key[i] = -1e30f;
            val[i] = 0;
        }
    }
    __syncthreads();

    // ascending bitonic sort; output read back reversed => descending top-K
    for (int k2 = 2; k2 <= SORT_N; k2 <<= 1) {
        for (int j = k2 >> 1; j > 0; j >>= 1) {
            for (int i = tid; i < SORT_N; i += 1024) {
                int ixj = i ^ j;
                if (ixj > i) {
                    bool up = ((i & k2) == 0);
                    float ki = key[i], kx = key[ixj];
                    if ((ki > kx) == up) {
                        key[i] = kx; key[ixj] = ki;
                        int vi = val[i]; val[i] = val[ixj]; val[ixj] = vi;
                    }
                }
            }
            __syncthreads();
        }
    }

    const float* rb = reg + (size_t)n * 4 * HW;
    for (int k = tid; k < K_TOP; k += 1024) {
        int src = SORT_N - 1 - k;
        float s = key[src];
        int idx = val[src];
        int loc = idx / C_CLS;
        int c   = idx - loc * C_CLS;

        float4 a = *(const float4*)(anchors + (size_t)loc * 4);
        float w  = a.z - a.x + 1.0f;
        float h  = a.w - a.y + 1.0f;
        float cx = a.x + 0.5f * w;
        float cy = a.y + 0.5f * h;

        float dx = rb[0 * HW + loc] / 10.0f;
        float dy = rb[1 * HW + loc] / 10.0f;
        float dw = fminf(rb[2 * HW + loc] / 5.0f, BBOX_CLIP);
        float dh = fminf(rb[3 * HW + loc] / 5.0f, BBOX_CLIP);

        float pcx = dx * w + cx;
        float pcy = dy * h + cy;
        float pw  = __expf(dw) * w;
        float ph  = __expf(dh) * h;

        float x1 = pcx - 0.5f * pw;
        float y1 = pcy - 0.5f * ph;
        float x2 = pcx + 0.5f * pw - 1.0f;
        float y2 = pcy + 0.5f * ph - 1.0f;

        x1 = fminf(fmaxf(x1, 0.0f), IMG_W_M1);
        y1 = fminf(fmaxf(y1, 0.0f), IMG_H_M1);
        x2 = fminf(fmaxf(x2, 0.0f), IMG_W_M1);
        y2 = fminf(fmaxf(y2, 0.0f), IMG_H_M1);

        bool keep = (x2 - x1 + 1.0f >= 0.0f) && (y2 - y1 + 1.0f >= 0.0f) && (s > 0.0f);
        float so  = keep ? s : 0.0f;

        float* o = out + ((size_t)n * K_TOP + k) * 5;
        o[0] = x1; o[1] = y1; o[2] = x2; o[3] = y2; o[4] = so;
        // labels (second tuple output), stored as float after the [N,K,5] block
        out[(size_t)N_IMG * K_TOP * 5 + (size_t)n * K_TOP + k] = (float)(c + 1);
    }
}

// ---------------- launcher ----------------
extern "C" void kernel_launch(void* const* d_in, const int* in_sizes, int n_in,
                              void* d_out, int out_size, void* d_ws, size_t ws_size,
                              hipStream_t stream)
{
    (void)in_sizes; (void)n_in; (void)out_size; (void)ws_size;
    const float* cls     = (const float*)d_in[0];   // [N, C, H, W]
    const float* reg     = (const float*)d_in[1];   // [N, 4, H, W]
    const float* iou     = (const float*)d_in[2];   // [N, 1, H, W]
    const float* anchors = (const float*)d_in[3];   // [HW, 4]
    float* out = (float*)d_out;                     // [N,K,5] ++ labels [N,K]

    char* ws = (char*)d_ws;
    size_t off = 0;
    unsigned* hist = (unsigned*)(ws + off); off += (size_t)N_IMG * BINS * 4;   // 256 KB
    unsigned* cnt  = (unsigned*)(ws + off); off += (size_t)N_IMG * 2 * 4;
    int*      meta = (int*)     (ws + off); off += (size_t)N_IMG * 4 * 4;
    float*  cscore = (float*)   (ws + off); off += (size_t)N_IMG * K_TOP * 4;  // 320 KB
    int*      cidx = (int*)     (ws + off);                                    // 320 KB

    // hist and cnt are contiguous: zero both in one shot (every call -> deterministic)
    const int zn = N_IMG * BINS + N_IMG * 2;
    k_zero<<<(zn + 255) / 256, 256, 0, stream>>>(hist, zn);

    dim3 g1(TILES, N_IMG);
    k_hist<<<g1, 256, 0, stream>>>(cls, iou, hist);

    k_select<<<N_IMG, 32, 0, stream>>>(hist, meta);

    dim3 g3(TILES, N_IMG);
    k_compact<<<g3, 256, 0, stream>>>(cls, iou, meta, cnt, cscore, cidx);

    k_sort_out<<<N_IMG, 1024, (size_t)SORT_N * 8, stream>>>(
        reg, anchors, meta, cscore, cidx, out);
}